// global_pool_50981261804240
// MI455X (gfx1250) — compile-verified
//
#include <hip/hip_runtime.h>
#include <math.h>

#define NHID 256
#define BGRAPHS 4096

typedef float v2f __attribute__((ext_vector_type(2)));
typedef float v8f __attribute__((ext_vector_type(8)));

// ---------------------------------------------------------------------------
// Kernel 1: exclusive scan of c_size -> off[B+1]   (single block, 256 threads)
// ---------------------------------------------------------------------------
__global__ void scan_offsets_kernel(const int* __restrict__ c_size,
                                    int* __restrict__ off) {
  __shared__ int tot[256];
  const int t = threadIdx.x;
  constexpr int ITEMS = BGRAPHS / 256;
  const int base = t * ITEMS;
  int vals[ITEMS];
  int s = 0;
#pragma unroll
  for (int i = 0; i < ITEMS; ++i) { vals[i] = c_size[base + i]; s += vals[i]; }
  tot[t] = s;
  __syncthreads();
  for (int d = 1; d < 256; d <<= 1) {
    int addv = (t >= d) ? tot[t - d] : 0;
    __syncthreads();
    tot[t] += addv;
    __syncthreads();
  }
  int run = tot[t] - s;  // exclusive prefix of this thread's chunk
#pragma unroll
  for (int i = 0; i < ITEMS; ++i) { off[base + i] = run; run += vals[i]; }
  if (t == 255) off[BGRAPHS] = run;
}

// ---------------------------------------------------------------------------
// Kernel 2: per-graph segment mean.  grid = B blocks, 256 threads.
// Block = 4 row-subgroups x 64 threads; each thread owns 4 contiguous
// features (one float4 / global_load_b128 per row), LDS-reduced across
// subgroups, stored as one b128 per thread.
// ---------------------------------------------------------------------------
__global__ void segment_mean_kernel(const float* __restrict__ x,
                                    const int* __restrict__ off,
                                    const int* __restrict__ c_size,
                                    float* __restrict__ mean) {
  __shared__ float4 red[256];
  const int b   = blockIdx.x;
  const int t   = threadIdx.x;
  const int sub = t >> 6;          // row subgroup 0..3
  const int fg  = (t & 63) * 4;    // feature base (16B aligned)
  const int s = off[b];
  const int e = off[b + 1];

  float ax = 0.f, ay = 0.f, az = 0.f, aw = 0.f;
  for (int r = s + sub; r < e; r += 4) {
    const float4 v = *(const float4*)(x + (size_t)r * NHID + fg);
    ax += v.x; ay += v.y; az += v.z; aw += v.w;
  }
  red[t] = make_float4(ax, ay, az, aw);
  __syncthreads();
  if (sub == 0) {
    const float4 v1 = red[t + 64];
    const float4 v2 = red[t + 128];
    const float4 v3 = red[t + 192];
    const float inv = 1.f / (float)max(c_size[b], 1);
    float4 o;
    o.x = (ax + v1.x + v2.x + v3.x) * inv;
    o.y = (ay + v1.y + v2.y + v3.y) * inv;
    o.z = (az + v1.z + v2.z + v3.z) * inv;
    o.w = (aw + v1.w + v2.w + v3.w) * inv;
    *(float4*)(mean + (size_t)b * NHID + fg) = o;
  }
}

// ---------------------------------------------------------------------------
// Kernel 3: c = tanh(mean @ W) via V_WMMA_F32_16X16X4_F32.
// 256 threads = 8 waves/block; each wave computes one 16x16 tile of c,
// looping K=NHID in steps of 4 (64 WMMAs per wave).
// A (16x4 f32): lanes 0-15 = rows M, VGPR0/1 = K pair; half-wave selects K+2.
// B (4x16 f32): lanes = cols N, VGPR0/1 = K pair; half-wave selects K+2.
// C/D (16x16 f32): VGPR g -> row g (lanes 0-15) / row g+8 (lanes 16-31).
// ---------------------------------------------------------------------------
__global__ void context_gemm_tanh_kernel(const float* __restrict__ mean,
                                         const float* __restrict__ W,
                                         float* __restrict__ ctx) {
  const int lane = threadIdx.x & 31;
  const int wave = threadIdx.x >> 5;
  const int half = lane >> 4;   // 0: K+0/K+1, 1: K+2/K+3
  const int l    = lane & 15;

  const int tile   = blockIdx.x * 8 + wave;     // 4096 tiles total
  const int tilesN = NHID / 16;                 // 16
  const int tm     = (tile / tilesN) * 16;      // row tile base
  const int tn     = (tile % tilesN) * 16;      // col tile base

  v8f acc = {};
  const float* arow = mean + (size_t)(tm + l) * NHID;
  for (int k = 0; k < NHID; k += 4) {
    const int ka = k + 2 * half;
    v2f a, bm;
    a.x  = arow[ka + 0];
    a.y  = arow[ka + 1];
    bm.x = W[(size_t)(ka + 0) * NHID + tn + l];
    bm.y = W[(size_t)(ka + 1) * NHID + tn + l];
    acc = __builtin_amdgcn_wmma_f32_16x16x4_f32(
        /*neg_a=*/false, a, /*neg_b=*/false, bm,
        /*c_mod=*/(short)0, acc, /*reuse_a=*/false, /*reuse_b=*/false);
  }
#pragma unroll
  for (int g = 0; g < 8; ++g) {
    const int row = tm + half * 8 + g;
    ctx[(size_t)row * NHID + tn + l] = tanhf(acc[g]);
  }
}

// ---------------------------------------------------------------------------
// Kernel 4: per-node gate + segment sum.  grid = B blocks, 256 threads.
// ctx[b] cached in LDS then hoisted into registers; 8 waves take rows
// round-robin; each lane reads its row slice as two float4s (b128, 512B
// contiguous per wave per load); wave32 shuffle reduction for the dot;
// alpha*x accumulated in registers, combined across waves via ds_add_f32.
// ---------------------------------------------------------------------------
__global__ void gate_pool_kernel(const float* __restrict__ x,
                                 const int* __restrict__ off,
                                 const float* __restrict__ ctx,
                                 float* __restrict__ out) {
  __shared__ float c_sh[NHID];
  __shared__ float acc_sh[NHID];
  const int b = blockIdx.x;
  const int t = threadIdx.x;
  c_sh[t]   = ctx[(size_t)b * NHID + t];
  acc_sh[t] = 0.f;
  __syncthreads();

  const int s = off[b];
  const int e = off[b + 1];
  const int wave = t >> 5;
  const int lane = t & 31;
  const int f0 = lane * 4;        // features f0..f0+3
  const int f1 = 128 + lane * 4;  // features f1..f1+3

  float c0[4], c1[4];
#pragma unroll
  for (int j = 0; j < 4; ++j) { c0[j] = c_sh[f0 + j]; c1[j] = c_sh[f1 + j]; }

  float r0[4] = {0.f, 0.f, 0.f, 0.f};
  float r1[4] = {0.f, 0.f, 0.f, 0.f};

  for (int r = s + wave; r < e; r += 8) {
    const float* xr = x + (size_t)r * NHID;
    const float4 xa = *(const float4*)(xr + f0);
    const float4 xb = *(const float4*)(xr + f1);
    float dot = xa.x * c0[0] + xa.y * c0[1] + xa.z * c0[2] + xa.w * c0[3]
              + xb.x * c1[0] + xb.y * c1[1] + xb.z * c1[2] + xb.w * c1[3];
#pragma unroll
    for (int m = 16; m >= 1; m >>= 1) dot += __shfl_xor(dot, m, 32);
    const float alpha = 1.f / (1.f + expf(-dot));
    r0[0] += alpha * xa.x; r0[1] += alpha * xa.y;
    r0[2] += alpha * xa.z; r0[3] += alpha * xa.w;
    r1[0] += alpha * xb.x; r1[1] += alpha * xb.y;
    r1[2] += alpha * xb.z; r1[3] += alpha * xb.w;
  }

#pragma unroll
  for (int j = 0; j < 4; ++j) {
    atomicAdd(&acc_sh[f0 + j], r0[j]);
    atomicAdd(&acc_sh[f1 + j], r1[j]);
  }
  __syncthreads();
  out[(size_t)b * NHID + t] = acc_sh[t];
}

// ---------------------------------------------------------------------------
extern "C" void kernel_launch(void* const* d_in, const int* in_sizes, int n_in,
                              void* d_out, int out_size, void* d_ws, size_t ws_size,
                              hipStream_t stream) {
  const float* x      = (const float*)d_in[0];   // [N, 256]
  // d_in[1] = batch (sorted) -- offsets recomputed from c_size instead
  const int*   c_size = (const int*)d_in[2];     // [B]
  const float* W      = (const float*)d_in[3];   // [256, 256]
  float* out = (float*)d_out;                    // [B, 256]

  // Workspace layout (needs ~8.3 MB)
  unsigned char* ws = (unsigned char*)d_ws;
  int*   off  = (int*)ws;                                   // (B+1) ints
  float* mean = (float*)(ws + 32768);                       // B*256 f32
  float* ctx  = (float*)(ws + 32768 + (size_t)BGRAPHS * NHID * 4);

  scan_offsets_kernel<<<1, 256, 0, stream>>>(c_size, off);
  segment_mean_kernel<<<BGRAPHS, 256, 0, stream>>>(x, off, c_size, mean);
  const int tiles = (BGRAPHS / 16) * (NHID / 16);           // 4096
  context_gemm_tanh_kernel<<<tiles / 8, 256, 0, stream>>>(mean, W, ctx);
  gate_pool_kernel<<<BGRAPHS, 256, 0, stream>>>(x, off, ctx, out);
}